// AnomalyGNN_12893491822678
// MI455X (gfx1250) — compile-verified
//
#include <hip/hip_runtime.h>

#define IN_DIM 256
#define HID 128

typedef __attribute__((ext_vector_type(2))) float v2f;
typedef __attribute__((ext_vector_type(8))) float v8f;

// ---------------------------------------------------------------------------
// Degree / normalization
// ---------------------------------------------------------------------------
__global__ void k_deg_init(unsigned* __restrict__ deg, int N) {
    int i = blockIdx.x * blockDim.x + threadIdx.x;
    if (i < N) deg[i] = 1u;  // self-loop
}

__global__ void k_deg_count(const int* __restrict__ col, unsigned* __restrict__ deg, int E) {
    int i = blockIdx.x * blockDim.x + threadIdx.x;
    if (i < E) atomicAdd(&deg[col[i]], 1u);
}

__global__ void k_dinv(const unsigned* __restrict__ deg, float* __restrict__ dinv, int N) {
    int i = blockIdx.x * blockDim.x + threadIdx.x;
    if (i < N) dinv[i] = rsqrtf((float)deg[i]);
}

// ---------------------------------------------------------------------------
// Encoder GEMM: h[N,128] = x[N,256] @ W_enc[128,256]^T   (FP32 WMMA 16x16x4)
// Block: 256 threads = 8 waves; block tile 16 rows x 128 cols; wave owns a
// 16x16 tile; K tiled by 64 through LDS.
// ---------------------------------------------------------------------------
__launch_bounds__(256)
__global__ void k_enc_gemm(const float* __restrict__ x, const float* __restrict__ W,
                           float* __restrict__ h, int N) {
    __shared__ float sX[16][68];    // pad 4: float4 stores aligned, conflict-free b64 reads
    __shared__ float sW[128][68];

    const int tid  = threadIdx.x;
    const int wave = tid >> 5;
    const int lane = tid & 31;
    const int lo   = lane & 15;
    const int hi   = lane >> 4;        // 0: K={k,k+1}, 1: K={k+2,k+3}
    const int row0 = blockIdx.x * 16;
    const int col0 = wave * 16;

    v8f acc = {};

    for (int k0 = 0; k0 < IN_DIM; k0 += 64) {
        {   // X tile: 16x64 = 256 float4, one per thread
            int r  = tid >> 4;
            int c4 = (tid & 15) << 2;
            int rr = row0 + r; if (rr >= N) rr = N - 1;
            float4 v = *(const float4*)(x + (size_t)rr * IN_DIM + k0 + c4);
            *(float4*)&sX[r][c4] = v;
        }
        // W tile: 128x64 = 2048 float4, 8 per thread (W is L2-resident, 128 KB)
        #pragma unroll
        for (int i = 0; i < 8; ++i) {
            int f4 = tid + i * 256;
            int r  = f4 >> 4;
            int c4 = (f4 & 15) << 2;
            float4 v = *(const float4*)(W + (size_t)r * IN_DIM + k0 + c4);
            *(float4*)&sW[r][c4] = v;
        }
        __syncthreads();

        #pragma unroll
        for (int kk = 0; kk < 64; kk += 4) {
            v2f a = *(const v2f*)&sX[lo][kk + 2 * hi];
            v2f b = *(const v2f*)&sW[col0 + lo][kk + 2 * hi];
            acc = __builtin_amdgcn_wmma_f32_16x16x4_f32(false, a, false, b,
                                                        (short)0, acc, false, false);
        }
        __syncthreads();
    }

    // D layout: VGPR j -> lanes 0-15: M=j, lanes 16-31: M=j+8
    #pragma unroll
    for (int j = 0; j < 8; ++j) {
        int m = row0 + j + 8 * hi;
        if (m < N) h[(size_t)m * HID + col0 + lo] = acc[j];
    }
}

// ---------------------------------------------------------------------------
// Seed accumulator with self-loop term: agg[n] = h[n] * dinv[n]^2
// ---------------------------------------------------------------------------
__global__ void k_selfloop(const float* __restrict__ h, const float* __restrict__ dinv,
                           float* __restrict__ agg, int N) {
    int idx = blockIdx.x * blockDim.x + threadIdx.x;   // N*32 float4 slots
    int n = idx >> 5;
    if (n >= N) return;
    int c4 = (idx & 31) << 2;
    float w = dinv[n] * dinv[n];
    float4 v = *(const float4*)(h + (size_t)n * HID + c4);
    v.x *= w; v.y *= w; v.z *= w; v.w *= w;
    *(float4*)(agg + (size_t)n * HID + c4) = v;
}

// ---------------------------------------------------------------------------
// Edge scatter: one wave per edge; gather 512B row of h, scale, atomic-add
// ---------------------------------------------------------------------------
__launch_bounds__(256)
__global__ void k_scatter(const int* __restrict__ rowi, const int* __restrict__ coli,
                          const float* __restrict__ h, const float* __restrict__ dinv,
                          float* __restrict__ agg, int E) {
    int e = blockIdx.x * 8 + (threadIdx.x >> 5);
    if (e >= E) return;
    int lane = threadIdx.x & 31;
    int r = rowi[e];
    int c = coli[e];
    float w = dinv[r] * dinv[c];
    float4 m = *(const float4*)(h + (size_t)r * HID + lane * 4);
    float* dst = agg + (size_t)c * HID + lane * 4;
    unsafeAtomicAdd(dst + 0, m.x * w);
    unsafeAtomicAdd(dst + 1, m.y * w);
    unsafeAtomicAdd(dst + 2, m.z * w);
    unsafeAtomicAdd(dst + 3, m.w * w);
}

// ---------------------------------------------------------------------------
// z = relu(agg + b_enc), in place in d_out
// ---------------------------------------------------------------------------
__global__ void k_z_final(float* __restrict__ z, const float* __restrict__ b, long long total) {
    long long i = (long long)blockIdx.x * blockDim.x + threadIdx.x;
    if (i >= total) return;
    float v = z[i] + b[(int)(i & (HID - 1))];
    z[i] = v > 0.0f ? v : 0.0f;
}

// ---------------------------------------------------------------------------
// Decoder GEMM: xr[N,256] = z[N,128] @ W_dec[256,128]^T + b_dec
// Block tile 16 rows x 256 cols; 8 waves x 2 col-tiles; K tiled by 32.
// ---------------------------------------------------------------------------
__launch_bounds__(256)
__global__ void k_dec_gemm(const float* __restrict__ z, const float* __restrict__ W,
                           const float* __restrict__ bias, float* __restrict__ xr, int N) {
    __shared__ float sZ[16][36];
    __shared__ float sW[256][36];

    const int tid  = threadIdx.x;
    const int wave = tid >> 5;
    const int lane = tid & 31;
    const int lo   = lane & 15;
    const int hi   = lane >> 4;
    const int row0 = blockIdx.x * 16;
    const int colA = wave * 32;

    v8f c0 = {}, c1 = {};

    for (int k0 = 0; k0 < HID; k0 += 32) {
        if (tid < 128) {   // Z tile: 16x32 = 128 float4
            int r  = tid >> 3;
            int c4 = (tid & 7) << 2;
            int rr = row0 + r; if (rr >= N) rr = N - 1;
            *(float4*)&sZ[r][c4] = *(const float4*)(z + (size_t)rr * HID + k0 + c4);
        }
        #pragma unroll
        for (int i = 0; i < 8; ++i) {   // W tile: 256x32 = 2048 float4
            int f4 = tid + i * 256;
            int r  = f4 >> 3;
            int c4 = (f4 & 7) << 2;
            *(float4*)&sW[r][c4] = *(const float4*)(W + (size_t)r * HID + k0 + c4);
        }
        __syncthreads();

        #pragma unroll
        for (int kk = 0; kk < 32; kk += 4) {
            v2f a  = *(const v2f*)&sZ[lo][kk + 2 * hi];
            v2f b0 = *(const v2f*)&sW[colA + lo][kk + 2 * hi];
            v2f b1 = *(const v2f*)&sW[colA + 16 + lo][kk + 2 * hi];
            c0 = __builtin_amdgcn_wmma_f32_16x16x4_f32(false, a, false, b0,
                                                       (short)0, c0, false, false);
            c1 = __builtin_amdgcn_wmma_f32_16x16x4_f32(false, a, false, b1,
                                                       (short)0, c1, false, false);
        }
        __syncthreads();
    }

    #pragma unroll
    for (int j = 0; j < 8; ++j) {
        int m = row0 + j + 8 * hi;
        if (m < N) {
            xr[(size_t)m * IN_DIM + colA + lo]      = c0[j] + bias[colA + lo];
            xr[(size_t)m * IN_DIM + colA + 16 + lo] = c1[j] + bias[colA + 16 + lo];
        }
    }
}

// ---------------------------------------------------------------------------
extern "C" void kernel_launch(void* const* d_in, const int* in_sizes, int n_in,
                              void* d_out, int out_size, void* d_ws, size_t ws_size,
                              hipStream_t stream) {
    const float* x     = (const float*)d_in[0];
    const int*   ei    = (const int*)d_in[1];      // [2, E] row-major, int32
    const float* W_enc = (const float*)d_in[2];
    const float* b_enc = (const float*)d_in[3];
    const float* W_dec = (const float*)d_in[4];
    const float* b_dec = (const float*)d_in[5];

    const int N = in_sizes[0] / IN_DIM;
    const int E = in_sizes[1] / 2;
    const int* rowi = ei;        // sources
    const int* coli = ei + E;    // targets

    unsigned* deg  = (unsigned*)d_ws;
    float*    dinv = (float*)((char*)d_ws + (size_t)N * 4);
    float*    h    = (float*)((char*)d_ws + (size_t)2 * N * 4);

    float* z  = (float*)d_out;                       // [N, HID]
    float* xr = (float*)d_out + (size_t)N * HID;     // [N, IN_DIM]

    k_deg_init <<<(N + 255) / 256, 256, 0, stream>>>(deg, N);
    k_deg_count<<<(E + 255) / 256, 256, 0, stream>>>(coli, deg, E);
    k_dinv     <<<(N + 255) / 256, 256, 0, stream>>>(deg, dinv, N);

    k_enc_gemm <<<(N + 15) / 16, 256, 0, stream>>>(x, W_enc, h, N);

    k_selfloop <<<((size_t)N * 32 + 255) / 256, 256, 0, stream>>>(h, dinv, z, N);
    k_scatter  <<<(E + 7) / 8, 256, 0, stream>>>(rowi, coli, h, dinv, z, E);

    long long ztot = (long long)N * HID;
    k_z_final  <<<(unsigned)((ztot + 255) / 256), 256, 0, stream>>>(z, b_enc, ztot);

    k_dec_gemm <<<(N + 15) / 16, 256, 0, stream>>>(z, W_dec, b_dec, xr, N);
}